// RGAT_82497731822008
// MI455X (gfx1250) — compile-verified
//
#include <hip/hip_runtime.h>
#include <math.h>

#define N_NODES  50000   // multiple of 16 -> no tail guards needed in WMMA kernel
#define N_EDGES  800000
#define DIM_IN   16
#define DIM_HID  128
#define DIM_OUT  3
#define N_REL    2
#define NEG_SLOPE 0.2f

typedef __attribute__((ext_vector_type(2))) float v2f;
typedef __attribute__((ext_vector_type(8))) float v8f;

// ---------------------------------------------------------------------------
// float atomic max via signed/unsigned integer ordering trick.
// Memory must be initialized to -inf (0xFF800000).
// ---------------------------------------------------------------------------
__device__ __forceinline__ void atomic_max_float(float* addr, float val) {
  if (val >= 0.0f) {
    atomicMax((int*)addr, __float_as_int(val));
  } else {
    atomicMin((unsigned int*)addr, __float_as_uint(val));
  }
}

// ---------------------------------------------------------------------------
// Layer 1 node transform: xr1[r][n][c] = sum_k x[n][k] * W1[r][k][c]
// One wave computes one 16(node) x 16(col) tile via 4 chained
// V_WMMA_F32_16X16X4_F32 (K = 16 = 4 steps of 4).
// Block: 256 threads = 8 waves -> 8 column tiles = full HID=128.
// Grid: (N/16, R). N is a multiple of 16, so all accesses unconditional
// (EXEC stays all-ones around the WMMA chain, per ISA requirement).
//
// Operand layouts per CDNA5 ISA 7.12.2 (32-bit, wave32):
//   A 16x4 : lane<16 -> M=lane, half=lane>>4 ; v0 = A[M][2h], v1 = A[M][2h+1]
//   B 4x16 : lane<16 -> N=lane ; v0 = B[2h][N], v1 = B[2h+1][N]
//   D 16x16: vgpr v  -> row (v + 8*half), col = lane&15
// ---------------------------------------------------------------------------
__global__ __launch_bounds__(256) void k_transform1(
    const float* __restrict__ x,    // [N, 16]
    const float* __restrict__ W1,   // [R, 16, 128]
    float* __restrict__ xr1)        // [R, N, 128]
{
  const int tile = blockIdx.x;
  const int r    = blockIdx.y;
  const int wave = threadIdx.x >> 5;   // column tile 0..7
  const int lane = threadIdx.x & 31;
  const int mrow = lane & 15;          // M (for A) / N (for B,D) within tile
  const int half = lane >> 4;
  const int node = tile * 16 + mrow;   // A-matrix row
  const int col  = wave * 16 + mrow;   // B/D column

  // Hoist all A fragments: per lane, pairs (k, k+1) at k = 4s + 2*half.
  const float* xrow = x + (size_t)node * DIM_IN + half * 2;
  v2f a0 = *(const v2f*)(xrow + 0);
  v2f a1 = *(const v2f*)(xrow + 4);
  v2f a2 = *(const v2f*)(xrow + 8);
  v2f a3 = *(const v2f*)(xrow + 12);

  const float* wbase = W1 + ((size_t)r * DIM_IN + half * 2) * DIM_HID + col;
  v8f acc = {};
#pragma unroll
  for (int s = 0; s < 4; ++s) {
    const v2f a = (s == 0) ? a0 : (s == 1) ? a1 : (s == 2) ? a2 : a3;
    const float* wp = wbase + (size_t)s * 4 * DIM_HID;
    v2f b;
    b.x = wp[0];
    b.y = wp[DIM_HID];
    acc = __builtin_amdgcn_wmma_f32_16x16x4_f32(
        /*neg_a=*/false, a, /*neg_b=*/false, b,
        /*c_mod=*/(short)0, acc, /*reuse_a=*/false, /*reuse_b=*/false);
  }

  float* out = xr1 + ((size_t)r * N_NODES + tile * 16 + half * 8) * DIM_HID + col;
#pragma unroll
  for (int v = 0; v < 8; ++v) {
    out[(size_t)v * DIM_HID] = acc[v];
  }
}

// ---------------------------------------------------------------------------
// Node-level attention terms for layer 1: qi[r*N+n] = xr1[r][n][:] . q1
// ---------------------------------------------------------------------------
__global__ void k_scores1(const float* __restrict__ xr1,
                          const float* __restrict__ q1,
                          const float* __restrict__ k1,
                          float* __restrict__ qi,
                          float* __restrict__ kj)
{
  const int idx = blockIdx.x * blockDim.x + threadIdx.x;  // over R*N
  if (idx >= N_REL * N_NODES) return;
  const float* row = xr1 + (size_t)idx * DIM_HID;
  float aq = 0.0f, ak = 0.0f;
#pragma unroll 8
  for (int c = 0; c < DIM_HID; ++c) {
    const float xv = row[c];
    aq = fmaf(xv, q1[c], aq);
    ak = fmaf(xv, k1[c], ak);
  }
  qi[idx] = aq;
  kj[idx] = ak;
}

// ---------------------------------------------------------------------------
// Init per-node softmax state: max = -inf, denom = 0
// ---------------------------------------------------------------------------
__global__ void k_init_nodes(float* __restrict__ maxb, float* __restrict__ denom)
{
  const int n = blockIdx.x * blockDim.x + threadIdx.x;
  if (n < N_NODES) {
    ((unsigned int*)maxb)[n] = 0xFF800000u;  // -inf
    denom[n] = 0.0f;
  }
}

// Init accumulator buffer to broadcast bias (out = segment_sum(...) + b)
__global__ void k_init_bias(float* __restrict__ out, const float* __restrict__ bias,
                            int nfeat, int total)
{
  const int i = blockIdx.x * blockDim.x + threadIdx.x;
  if (i < total) out[i] = bias[i % nfeat];
}

// ---------------------------------------------------------------------------
// Per-edge attention logit + segment max (atomic)
// alpha[e] = leaky_relu(qi[et,dst] + kj[et,src])
// ---------------------------------------------------------------------------
__global__ void k_edge_alpha(const int* __restrict__ src, const int* __restrict__ dst,
                             const int* __restrict__ et,
                             const float* __restrict__ qi, const float* __restrict__ kj,
                             float* __restrict__ alpha, float* __restrict__ maxb)
{
  const int e = blockIdx.x * blockDim.x + threadIdx.x;
  if (e >= N_EDGES) return;
  const int r = et[e], s = src[e], d = dst[e];
  float al = qi[r * N_NODES + d] + kj[r * N_NODES + s];
  al = (al >= 0.0f) ? al : NEG_SLOPE * al;
  alpha[e] = al;
  atomic_max_float(maxb + d, al);
}

// Segment sum of exp(alpha - max[dst])
__global__ void k_edge_denom(const int* __restrict__ dst,
                             const float* __restrict__ alpha,
                             const float* __restrict__ maxb,
                             float* __restrict__ denom)
{
  const int e = blockIdx.x * blockDim.x + threadIdx.x;
  if (e >= N_EDGES) return;
  const int d = dst[e];
  atomicAdd(denom + d, __expf(alpha[e] - maxb[d]));
}

// ---------------------------------------------------------------------------
// Layer-1 aggregation: h[dst][c] += a_e * xr1[et][src][c]
// One wave (32 lanes) per edge; each lane handles 4 channels via float4.
// Block = 256 threads = 8 edges.
// ---------------------------------------------------------------------------
__global__ __launch_bounds__(256) void k_aggregate1(
    const int* __restrict__ src, const int* __restrict__ dst,
    const int* __restrict__ et,
    const float* __restrict__ alpha, const float* __restrict__ maxb,
    const float* __restrict__ denom,
    const float* __restrict__ xr1, float* __restrict__ h)
{
  const int e = blockIdx.x * 8 + (threadIdx.x >> 5);
  if (e >= N_EDGES) return;
  const int lane = threadIdx.x & 31;
  const int c = lane * 4;
  const int r = et[e], s = src[e], d = dst[e];
  const float a = __expf(alpha[e] - maxb[d]) / (denom[d] + 1e-16f);
  const float4 xv = *(const float4*)(xr1 + ((size_t)r * N_NODES + s) * DIM_HID + c);
  float* hp = h + (size_t)d * DIM_HID + c;
  atomicAdd(hp + 0, a * xv.x);
  atomicAdd(hp + 1, a * xv.y);
  atomicAdd(hp + 2, a * xv.z);
  atomicAdd(hp + 3, a * xv.w);
}

// ---------------------------------------------------------------------------
// Layer 2 node transform + attention terms (OUT=3, scalar path)
// ---------------------------------------------------------------------------
__global__ void k_transform2(const float* __restrict__ h,   // [N, 128]
                             const float* __restrict__ W2,  // [R, 128, 3]
                             const float* __restrict__ q2,  // [3]
                             const float* __restrict__ k2,  // [3]
                             float* __restrict__ xr2,       // [R, N, 3]
                             float* __restrict__ qi,        // [R*N]
                             float* __restrict__ kj)        // [R*N]
{
  const int idx = blockIdx.x * blockDim.x + threadIdx.x;  // r*N + n
  if (idx >= N_REL * N_NODES) return;
  const int r = idx / N_NODES;
  const int n = idx - r * N_NODES;
  const float* row = h + (size_t)n * DIM_HID;
  const float* w   = W2 + (size_t)r * DIM_HID * DIM_OUT;
  float a0 = 0.0f, a1 = 0.0f, a2 = 0.0f;
#pragma unroll 8
  for (int c = 0; c < DIM_HID; ++c) {
    const float xv = row[c];
    a0 = fmaf(xv, w[c * 3 + 0], a0);
    a1 = fmaf(xv, w[c * 3 + 1], a1);
    a2 = fmaf(xv, w[c * 3 + 2], a2);
  }
  xr2[(size_t)idx * 3 + 0] = a0;
  xr2[(size_t)idx * 3 + 1] = a1;
  xr2[(size_t)idx * 3 + 2] = a2;
  qi[idx] = a0 * q2[0] + a1 * q2[1] + a2 * q2[2];
  kj[idx] = a0 * k2[0] + a1 * k2[1] + a2 * k2[2];
}

// Layer-2 aggregation: out[dst][0..2] += a_e * xr2[et][src][0..2]
__global__ void k_aggregate2(
    const int* __restrict__ src, const int* __restrict__ dst,
    const int* __restrict__ et,
    const float* __restrict__ alpha, const float* __restrict__ maxb,
    const float* __restrict__ denom,
    const float* __restrict__ xr2, float* __restrict__ out)
{
  const int e = blockIdx.x * blockDim.x + threadIdx.x;
  if (e >= N_EDGES) return;
  const int r = et[e], s = src[e], d = dst[e];
  const float a = __expf(alpha[e] - maxb[d]) / (denom[d] + 1e-16f);
  const float* xs = xr2 + ((size_t)r * N_NODES + s) * 3;
  atomicAdd(out + d * 3 + 0, a * xs[0]);
  atomicAdd(out + d * 3 + 1, a * xs[1]);
  atomicAdd(out + d * 3 + 2, a * xs[2]);
}

// ---------------------------------------------------------------------------
// Launch
// ---------------------------------------------------------------------------
extern "C" void kernel_launch(void* const* d_in, const int* in_sizes, int n_in,
                              void* d_out, int out_size, void* d_ws, size_t ws_size,
                              hipStream_t stream) {
  (void)in_sizes; (void)n_in; (void)out_size; (void)ws_size;

  const float* feature = (const float*)d_in[0];   // [N, 16]
  const int*   eidx    = (const int*)d_in[1];     // [2, E]
  const int*   etype   = (const int*)d_in[2];     // [E]
  const float* W1 = (const float*)d_in[3];        // [R, 16, 128]
  const float* q1 = (const float*)d_in[4];        // [128]
  const float* k1 = (const float*)d_in[5];        // [128]
  const float* b1 = (const float*)d_in[6];        // [128]
  const float* W2 = (const float*)d_in[7];        // [R, 128, 3]
  const float* q2 = (const float*)d_in[8];        // [3]
  const float* k2 = (const float*)d_in[9];        // [3]
  const float* b2 = (const float*)d_in[10];       // [3]
  float* out = (float*)d_out;                     // [N, 3]

  const int* src = eidx;            // row 0 = source j
  const int* dst = eidx + N_EDGES;  // row 1 = target i

  // Workspace layout (floats), ~83 MB total
  float* ws = (float*)d_ws;
  size_t off = 0;
  float* xr1   = ws + off; off += (size_t)N_REL * N_NODES * DIM_HID; // 12.8M
  float* qi1   = ws + off; off += (size_t)N_REL * N_NODES;
  float* kj1   = ws + off; off += (size_t)N_REL * N_NODES;
  float* alpha = ws + off; off += (size_t)N_EDGES;                   // reused by layer 2
  float* maxb  = ws + off; off += (size_t)N_NODES;                   // reused
  float* denom = ws + off; off += (size_t)N_NODES;                   // reused
  float* h     = ws + off; off += (size_t)N_NODES * DIM_HID;         // 6.4M
  float* xr2   = ws + off; off += (size_t)N_REL * N_NODES * DIM_OUT;
  float* qi2   = ws + off; off += (size_t)N_REL * N_NODES;
  float* kj2   = ws + off; off += (size_t)N_REL * N_NODES;

  const int RN = N_REL * N_NODES;

  // ----- Layer 1 -----
  {
    dim3 grid(N_NODES / 16, N_REL);
    k_transform1<<<grid, 256, 0, stream>>>(feature, W1, xr1);
  }
  k_scores1<<<(RN + 255) / 256, 256, 0, stream>>>(xr1, q1, k1, qi1, kj1);
  k_init_nodes<<<(N_NODES + 255) / 256, 256, 0, stream>>>(maxb, denom);
  k_init_bias<<<((N_NODES * DIM_HID) + 255) / 256, 256, 0, stream>>>(
      h, b1, DIM_HID, N_NODES * DIM_HID);
  k_edge_alpha<<<(N_EDGES + 255) / 256, 256, 0, stream>>>(
      src, dst, etype, qi1, kj1, alpha, maxb);
  k_edge_denom<<<(N_EDGES + 255) / 256, 256, 0, stream>>>(dst, alpha, maxb, denom);
  k_aggregate1<<<(N_EDGES + 7) / 8, 256, 0, stream>>>(
      src, dst, etype, alpha, maxb, denom, xr1, h);

  // ----- Layer 2 -----
  k_transform2<<<(RN + 255) / 256, 256, 0, stream>>>(h, W2, q2, k2, xr2, qi2, kj2);
  k_init_nodes<<<(N_NODES + 255) / 256, 256, 0, stream>>>(maxb, denom);
  k_init_bias<<<((N_NODES * DIM_OUT) + 255) / 256, 256, 0, stream>>>(
      out, b2, DIM_OUT, N_NODES * DIM_OUT);
  k_edge_alpha<<<(N_EDGES + 255) / 256, 256, 0, stream>>>(
      src, dst, etype, qi2, kj2, alpha, maxb);
  k_edge_denom<<<(N_EDGES + 255) / 256, 256, 0, stream>>>(dst, alpha, maxb, denom);
  k_aggregate2<<<(N_EDGES + 255) / 256, 256, 0, stream>>>(
      src, dst, etype, alpha, maxb, denom, xr2, out);
}